// SofterMax_60808146977136
// MI455X (gfx1250) — compile-verified
//
#include <hip/hip_runtime.h>
#include <math.h>

// SofterMax v1 (base-2, online running-max WITHOUT denominator rescale).
// Bandwidth-bound: 512 MiB traffic @ 23.3 TB/s ~ 23 us floor.
// One 256-thread block (8 wave32s) per row of 4096 fp32 elements.
// Streaming loads/stores use non-temporal hints (TH=NT) -- no reuse, keep
// WGP$/L2 clean.

#define WAVE 32
constexpr int BLOCK  = 256;
constexpr int VLEN   = 4096;
constexpr int PT     = VLEN / BLOCK;   // 16 elements per thread (contiguous)
constexpr int PT4    = PT / 4;         // 4 x b128 per thread
constexpr int NWAVES = BLOCK / WAVE;   // 8 waves per block

typedef float v4f __attribute__((ext_vector_type(4)));

__global__ __launch_bounds__(BLOCK)
void softermax_kernel(const float* __restrict__ x,
                      float* __restrict__ y) {
    const int row = blockIdx.x;

    const v4f* __restrict__ xr =
        reinterpret_cast<const v4f*>(x + (size_t)row * VLEN);
    v4f* __restrict__ yr =
        reinterpret_cast<v4f*>(y + (size_t)row * VLEN);

    const int t    = threadIdx.x;
    const int lane = t & (WAVE - 1);
    const int wv   = t >> 5;

    // ---- 1) Load 16 contiguous floats (4 x b128, non-temporal) -----------
    v4f ld[PT4];
#pragma unroll
    for (int i = 0; i < PT4; ++i)
        ld[i] = __builtin_nontemporal_load(&xr[t * PT4 + i]);

    float a[PT];
#pragma unroll
    for (int i = 0; i < PT4; ++i) {
        a[4 * i + 0] = ld[i].x;
        a[4 * i + 1] = ld[i].y;
        a[4 * i + 2] = ld[i].z;
        a[4 * i + 3] = ld[i].w;
    }

    // ---- 2) Thread-chunk max, then wave32 inclusive max-scan -------------
    float cmax = a[0];
#pragma unroll
    for (int k = 1; k < PT; ++k) cmax = fmaxf(cmax, a[k]);

    float inc = cmax;
#pragma unroll
    for (int off = 1; off < WAVE; off <<= 1) {
        float o = __shfl_up(inc, (unsigned)off, WAVE);
        if (lane >= off) inc = fmaxf(inc, o);
    }
    float lexcl = __shfl_up(inc, 1u, WAVE);          // exclusive lane prefix
    if (lane == 0) lexcl = -__builtin_inff();

    // ---- 3) Cross-wave combine through LDS -------------------------------
    __shared__ float s_wmax[NWAVES];
    __shared__ float s_wsum[NWAVES];
    if (lane == WAVE - 1) s_wmax[wv] = inc;          // wave-inclusive max
    __syncthreads();

    float wexcl = -__builtin_inff();
    float mfin  = -__builtin_inff();
#pragma unroll
    for (int w = 0; w < NWAVES; ++w) {
        float wm = s_wmax[w];
        if (w < wv) wexcl = fmaxf(wexcl, wm);        // exclusive wave prefix
        mfin = fmaxf(mfin, wm);                      // row max
    }
    const float prefix = fmaxf(wexcl, lexcl);        // exclusive prefix max
                                                     // before this thread
    // ---- 4) Running max restart + denominator partial --------------------
    // m_run[k] = max(prefix, inclusive-scan-of-a[0..k]) == rm below.
    float rm   = prefix;
    float dsum = 0.0f;
#pragma unroll
    for (int k = 0; k < PT; ++k) {
        rm = fmaxf(rm, a[k]);
        dsum += __builtin_amdgcn_exp2f(a[k] - rm);   // v_exp_f32 (base-2)
    }

    // wave32 sum-reduce
#pragma unroll
    for (int off = WAVE / 2; off > 0; off >>= 1)
        dsum += __shfl_xor(dsum, off, WAVE);
    if (lane == 0) s_wsum[wv] = dsum;
    __syncthreads();

    float d = 0.0f;
#pragma unroll
    for (int w = 0; w < NWAVES; ++w) d += s_wsum[w];
    const float rinv = 1.0f / d;                     // d >= 1, safe

    // ---- 5) Normalize + store (4 x b128, non-temporal) -------------------
#pragma unroll
    for (int i = 0; i < PT4; ++i) {
        v4f o;
        o.x = __builtin_amdgcn_exp2f(a[4 * i + 0] - mfin) * rinv;
        o.y = __builtin_amdgcn_exp2f(a[4 * i + 1] - mfin) * rinv;
        o.z = __builtin_amdgcn_exp2f(a[4 * i + 2] - mfin) * rinv;
        o.w = __builtin_amdgcn_exp2f(a[4 * i + 3] - mfin) * rinv;
        __builtin_nontemporal_store(o, &yr[t * PT4 + i]);
    }
}

extern "C" void kernel_launch(void* const* d_in, const int* in_sizes, int n_in,
                              void* d_out, int out_size, void* d_ws, size_t ws_size,
                              hipStream_t stream) {
    const float* x = (const float*)d_in[0];
    float*       y = (float*)d_out;
    const int rows = in_sizes[0] / VLEN;             // 8*2048 = 16384 rows
    softermax_kernel<<<dim3(rows), dim3(BLOCK), 0, stream>>>(x, y);
}